// PCnet_24257975288186
// MI455X (gfx1250) — compile-verified
//
#include <hip/hip_runtime.h>
#include <hip/hip_bf16.h>
#include <math.h>

#define B_SZ 16
#define NPTS 1024
#define KNN  20

typedef float v2f __attribute__((ext_vector_type(2)));
typedef float v8f __attribute__((ext_vector_type(8)));

#define V8Z {0.f,0.f,0.f,0.f,0.f,0.f,0.f,0.f}

__device__ __forceinline__ v8f wmma4(v2f a, v2f b, v8f c) {
  // D = A(16x4 f32) x B(4x16 f32) + C(16x16 f32)
  return __builtin_amdgcn_wmma_f32_16x16x4_f32(false, a, false, b, (short)0, c, false, false);
}

// ---------------------------------------------------------------------------
// transpose (B,N,3) -> (B,3,N)
__global__ void k_transpose(const float* __restrict__ x, float* __restrict__ xc) {
  int t = blockIdx.x * 256 + threadIdx.x;
  if (t >= B_SZ * NPTS) return;
  int b = t / NPTS, n = t % NPTS;
  for (int c = 0; c < 3; ++c)
    xc[(size_t)b * 3 * NPTS + c * NPTS + n] = x[(size_t)b * NPTS * 3 + n * 3 + c];
}

// ---------------------------------------------------------------------------
// 3-dim kNN (VALU; K=3 cannot fill a WMMA): top-20 of -||xn-xm||^2, incl self.
template <int C>
__global__ __launch_bounds__(128) void k_knn(const float* __restrict__ xf, int* __restrict__ idx) {
  __shared__ float tile[32 * C];
  int b = blockIdx.y;
  int n = blockIdx.x * 128 + threadIdx.x;
  const float* xb = xf + (size_t)b * C * NPTS;
  float rowv[C];
#pragma unroll
  for (int c = 0; c < C; ++c) rowv[c] = xb[c * NPTS + n];
  float best[KNN];
  int bidx[KNN];
#pragma unroll
  for (int j = 0; j < KNN; ++j) { best[j] = -3.402823466e38f; bidx[j] = 0; }
  for (int m0 = 0; m0 < NPTS; m0 += 32) {
    __syncthreads();
    for (int i = threadIdx.x; i < 32 * C; i += 128) {
      int mm = i & 31, c = i >> 5;
      tile[c * 32 + mm] = xb[c * NPTS + m0 + mm];
    }
    __syncthreads();
    for (int mm = 0; mm < 32; ++mm) {
      float d = 0.f;
#pragma unroll
      for (int c = 0; c < C; ++c) { float t = tile[c * 32 + mm] - rowv[c]; d -= t * t; }
      if (d > best[KNN - 1]) {
        int j = KNN - 1;
        while (j > 0 && best[j - 1] < d) { best[j] = best[j - 1]; bidx[j] = bidx[j - 1]; --j; }
        best[j] = d; bidx[j] = m0 + mm;
      }
    }
  }
  for (int j = 0; j < KNN; ++j) idx[((size_t)b * NPTS + n) * KNN + j] = bidx[j];
}

// ---------------------------------------------------------------------------
// xx[b,n] = ||x[b,:,n]||^2 for C=64 feature maps (coalesced over n)
__global__ void k_sqnorm64(const float* __restrict__ xf, float* __restrict__ xx) {
  int t = blockIdx.x * 256 + threadIdx.x;
  if (t >= B_SZ * NPTS) return;
  int b = t / NPTS, n = t % NPTS;
  const float* xb = xf + (size_t)b * 64 * NPTS;
  float s = 0.f;
#pragma unroll 8
  for (int c = 0; c < 64; ++c) { float v = xb[c * NPTS + n]; s += v * v; }
  xx[t] = s;
}

// ---------------------------------------------------------------------------
// 64-dim kNN via WMMA: block = 8 waves handles 128 rows of batch b.
// Chunks of 32 candidate columns: dist tile = 2*(X^T X) - xx[m] computed on the
// matrix pipe, staged in LDS, then 1 compare/candidate top-20 scan (tid<128).
// Ordering identical to -||xn-xm||^2 (per-row constant -xx[n] dropped).
__global__ __launch_bounds__(256) void k_knn_wmma(
    const float* __restrict__ xf, const float* __restrict__ xx,
    int* __restrict__ idx) {
  __shared__ float SD[128 * 34];  // stride 34: conflict-free stores + scans
  int b = blockIdx.y;
  int n0 = blockIdx.x * 128;
  int tid = threadIdx.x, wave = tid >> 5, lane = tid & 31, laneM = lane & 15, hi = lane >> 4;
  const float* xb = xf + (size_t)b * 64 * NPTS;
  // resident A strip: 16 rows (n0+wave*16+laneM) x K=64
  int arow = n0 + wave * 16 + laneM;
  v2f a[16];
#pragma unroll
  for (int s = 0; s < 16; ++s) {
    int ka = 4 * s + hi * 2;
    a[s].x = xb[ka * NPTS + arow];
    a[s].y = xb[(ka + 1) * NPTS + arow];
  }
  float best[KNN];
  int bidx[KNN];
#pragma unroll
  for (int j = 0; j < KNN; ++j) { best[j] = -3.402823466e38f; bidx[j] = 0; }
  for (int m0 = 0; m0 < NPTS; m0 += 32) {
    __syncthreads();  // SD reuse vs previous scan
    for (int sub = 0; sub < 2; ++sub) {
      int mcol = m0 + sub * 16 + laneM;
      v8f acc = V8Z;
#pragma unroll
      for (int s = 0; s < 16; ++s) {
        int kb = 4 * s + hi * 2;
        v2f bv;
        bv.x = xb[kb * NPTS + mcol];
        bv.y = xb[(kb + 1) * NPTS + mcol];
        acc = wmma4(a[s], bv, acc);
      }
      float xxm = xx[(size_t)b * NPTS + mcol];
#pragma unroll
      for (int r = 0; r < 8; ++r) {
        int rr = wave * 16 + r + hi * 8;  // local row index
        SD[rr * 34 + sub * 16 + laneM] = 2.f * acc[r] - xxm;
      }
    }
    __syncthreads();
    if (tid < 128) {
      for (int j = 0; j < 32; ++j) {
        float d = SD[tid * 34 + j];
        if (d > best[KNN - 1]) {
          int t = KNN - 1;
          while (t > 0 && best[t - 1] < d) { best[t] = best[t - 1]; bidx[t] = bidx[t - 1]; --t; }
          best[t] = d; bidx[t] = m0 + j;
        }
      }
    }
  }
  if (tid < 128)
    for (int j = 0; j < KNN; ++j)
      idx[((size_t)b * NPTS + n0 + tid) * KNN + j] = bidx[j];
}

// ---------------------------------------------------------------------------
// Fused edge-conv: gather graph feature (2C x 80 cols for 4 points x 20 nbrs)
// into LDS, WMMA vs W (COUT x 2C), bn+relu, max over k -> (B,COUT,N)
template <int CIN2, int COUT>
__global__ __launch_bounds__(256) void k_edgeconv(
    const float* __restrict__ xin, const int* __restrict__ idx,
    const float* __restrict__ W, const float* __restrict__ Wb,
    const float* __restrict__ Wg, const float* __restrict__ We,
    float* __restrict__ out) {
  constexpr int C  = CIN2 / 2;
  constexpr int KP = (CIN2 + 3) & ~3;
  constexpr int NK = KP / 4;
  constexpr int MT = COUT / 16;
  __shared__ float F[KP * 80];
  int b = blockIdx.y, n0 = blockIdx.x * 4, tid = threadIdx.x;
  for (int i = tid; i < KP * 80; i += 256) {
    int col = i % 80, c = i / 80;
    int p = col / 20, kk = col % 20, n = n0 + p;
    float v = 0.f;
    if (c < C) {
      int nb = idx[((size_t)b * NPTS + n) * KNN + kk];
      v = xin[(size_t)b * C * NPTS + c * NPTS + nb] - xin[(size_t)b * C * NPTS + c * NPTS + n];
    } else if (c < CIN2) {
      v = xin[(size_t)b * C * NPTS + (c - C) * NPTS + n];
    }
    F[c * 80 + col] = v;
  }
  __syncthreads();
  int wave = tid >> 5, lane = tid & 31, laneM = lane & 15, hi = lane >> 4;
  for (int tg = wave; tg < MT * 4; tg += 8) {
    int p = tg / MT, mt = tg % MT;
    int row = mt * 16 + laneM;
    v2f a[NK];
#pragma unroll
    for (int s = 0; s < NK; ++s) {
      int ka = 4 * s + hi * 2;
      a[s].x = (ka     < CIN2) ? W[row * CIN2 + ka]     : 0.f;
      a[s].y = (ka + 1 < CIN2) ? W[row * CIN2 + ka + 1] : 0.f;
    }
    int col0  = p * 20 + laneM;
    int colL1 = 16 + laneM;
    int col1  = p * 20 + ((colL1 < 20) ? colL1 : 0);
    v8f acc0 = V8Z, acc1 = V8Z;
#pragma unroll
    for (int s = 0; s < NK; ++s) {
      int kb = 4 * s + hi * 2;
      v2f bv; bv.x = F[kb * 80 + col0]; bv.y = F[(kb + 1) * 80 + col0];
      acc0 = wmma4(a[s], bv, acc0);
    }
#pragma unroll
    for (int s = 0; s < NK; ++s) {
      int kb = 4 * s + hi * 2;
      v2f bv; bv.x = F[kb * 80 + col1]; bv.y = F[(kb + 1) * 80 + col1];
      acc1 = wmma4(a[s], bv, acc1);
    }
#pragma unroll
    for (int r = 0; r < 8; ++r) {
      int m = mt * 16 + r + hi * 8;
      float bi = Wb[m], gg = Wg[m], bo = We[m];
      float v0 = fmaxf((acc0[r] + bi) * gg + bo, 0.f);
      float v1 = (laneM < 4) ? fmaxf((acc1[r] + bi) * gg + bo, 0.f) : 0.f;
      float v = fmaxf(v0, v1);
#pragma unroll
      for (int off = 1; off < 16; off <<= 1) v = fmaxf(v, __shfl_xor(v, off, 32));
      if (laneM == 0) out[(size_t)b * COUT * NPTS + m * NPTS + n0 + p] = v;
    }
  }
}

// ---------------------------------------------------------------------------
// T-Net fused conv1(6->64) + conv2(64->128) + max over k  -> (B,128,N)
__global__ __launch_bounds__(256) void k_tnet12(
    const float* __restrict__ xc, const int* __restrict__ idx,
    const float* __restrict__ W1, const float* __restrict__ b1,
    const float* __restrict__ g1, const float* __restrict__ e1,
    const float* __restrict__ W2, const float* __restrict__ b2,
    const float* __restrict__ g2, const float* __restrict__ e2,
    float* __restrict__ out) {
  __shared__ float F[8 * 80];
  __shared__ float Y1[64 * 80];
  int b = blockIdx.y, n0 = blockIdx.x * 4, tid = threadIdx.x;
  for (int i = tid; i < 8 * 80; i += 256) {
    int col = i % 80, c = i / 80;
    int p = col / 20, kk = col % 20, n = n0 + p;
    float v = 0.f;
    if (c < 3) {
      int nb = idx[((size_t)b * NPTS + n) * KNN + kk];
      v = xc[(size_t)b * 3 * NPTS + c * NPTS + nb] - xc[(size_t)b * 3 * NPTS + c * NPTS + n];
    } else if (c < 6) {
      v = xc[(size_t)b * 3 * NPTS + (c - 3) * NPTS + n];
    }
    F[c * 80 + col] = v;
  }
  __syncthreads();
  int wave = tid >> 5, lane = tid & 31, laneM = lane & 15, hi = lane >> 4;
  // phase 1: 6 (pad 8) -> 64, write relu(bn(.)) into LDS Y1
  for (int tg = wave; tg < 16; tg += 8) {
    int p = tg >> 2, mt = tg & 3;
    int row = mt * 16 + laneM;
    v2f a0, a1;
    { int ka = 0 + hi * 2; a0.x = (ka < 6) ? W1[row * 6 + ka] : 0.f; a0.y = (ka + 1 < 6) ? W1[row * 6 + ka + 1] : 0.f; }
    { int ka = 4 + hi * 2; a1.x = (ka < 6) ? W1[row * 6 + ka] : 0.f; a1.y = (ka + 1 < 6) ? W1[row * 6 + ka + 1] : 0.f; }
    for (int sub = 0; sub < 2; ++sub) {
      int colL = sub * 16 + laneM;
      int col = p * 20 + ((colL < 20) ? colL : 0);
      v8f acc = V8Z;
      v2f bv;
      bv.x = F[(0 + hi * 2) * 80 + col]; bv.y = F[(1 + hi * 2) * 80 + col];
      acc = wmma4(a0, bv, acc);
      bv.x = F[(4 + hi * 2) * 80 + col]; bv.y = F[(5 + hi * 2) * 80 + col];
      acc = wmma4(a1, bv, acc);
      if (colL < 20) {
#pragma unroll
        for (int r = 0; r < 8; ++r) {
          int m = mt * 16 + r + hi * 8;
          float v = fmaxf((acc[r] + b1[m]) * g1[m] + e1[m], 0.f);
          Y1[m * 80 + p * 20 + colL] = v;
        }
      }
    }
  }
  __syncthreads();
  // phase 2: 64 -> 128, max over k
  for (int tg = wave; tg < 32; tg += 8) {
    int p = tg >> 3, mt = tg & 7;
    int row = mt * 16 + laneM;
    v2f a[16];
#pragma unroll
    for (int s = 0; s < 16; ++s) {
      int ka = 4 * s + hi * 2;
      a[s].x = W2[row * 64 + ka]; a[s].y = W2[row * 64 + ka + 1];
    }
    int col0  = p * 20 + laneM;
    int colL1 = 16 + laneM;
    int col1  = p * 20 + ((colL1 < 20) ? colL1 : 0);
    v8f acc0 = V8Z, acc1 = V8Z;
#pragma unroll
    for (int s = 0; s < 16; ++s) {
      int kb = 4 * s + hi * 2;
      v2f bv; bv.x = Y1[kb * 80 + col0]; bv.y = Y1[(kb + 1) * 80 + col0];
      acc0 = wmma4(a[s], bv, acc0);
    }
#pragma unroll
    for (int s = 0; s < 16; ++s) {
      int kb = 4 * s + hi * 2;
      v2f bv; bv.x = Y1[kb * 80 + col1]; bv.y = Y1[(kb + 1) * 80 + col1];
      acc1 = wmma4(a[s], bv, acc1);
    }
#pragma unroll
    for (int r = 0; r < 8; ++r) {
      int m = mt * 16 + r + hi * 8;
      float bi = b2[m], gg = g2[m], bo = e2[m];
      float v0 = fmaxf((acc0[r] + bi) * gg + bo, 0.f);
      float v1 = (laneM < 4) ? fmaxf((acc1[r] + bi) * gg + bo, 0.f) : 0.f;
      float v = fmaxf(v0, v1);
#pragma unroll
      for (int off = 1; off < 16; off <<= 1) v = fmaxf(v, __shfl_xor(v, off, 32));
      if (laneM == 0) out[(size_t)b * 128 * NPTS + m * NPTS + n0 + p] = v;
    }
  }
}

// ---------------------------------------------------------------------------
// T-Net conv3: out[b,m] = max_n relu(bn(W(1024x128) @ y[b,:,n]))
__global__ __launch_bounds__(256) void k_gemm_maxN(
    const float* __restrict__ y, const float* __restrict__ W,
    const float* __restrict__ Wb, const float* __restrict__ Wg,
    const float* __restrict__ We, float* __restrict__ out) {
  int b = blockIdx.y;
  int tid = threadIdx.x, wave = tid >> 5, lane = tid & 31, laneM = lane & 15, hi = lane >> 4;
  int mbase = blockIdx.x * 128 + wave * 16;
  int row = mbase + laneM;
  v2f a[32];
#pragma unroll
  for (int s = 0; s < 32; ++s) {
    int ka = 4 * s + hi * 2;
    a[s].x = W[row * 128 + ka]; a[s].y = W[row * 128 + ka + 1];
  }
  float pb[8], pg[8], pe[8], vmax[8];
#pragma unroll
  for (int r = 0; r < 8; ++r) {
    int m = mbase + r + hi * 8;
    pb[r] = Wb[m]; pg[r] = Wg[m]; pe[r] = We[m]; vmax[r] = 0.f;
  }
  for (int n0 = 0; n0 < NPTS; n0 += 16) {
    v8f acc = V8Z;
#pragma unroll
    for (int s = 0; s < 32; ++s) {
      int kb = 4 * s + hi * 2;
      v2f bv;
      bv.x = y[(size_t)b * 128 * NPTS + kb * NPTS + n0 + laneM];
      bv.y = y[(size_t)b * 128 * NPTS + (kb + 1) * NPTS + n0 + laneM];
      acc = wmma4(a[s], bv, acc);
    }
#pragma unroll
    for (int r = 0; r < 8; ++r)
      vmax[r] = fmaxf(vmax[r], fmaxf((acc[r] + pb[r]) * pg[r] + pe[r], 0.f));
  }
#pragma unroll
  for (int r = 0; r < 8; ++r) {
    float v = vmax[r];
#pragma unroll
    for (int off = 1; off < 16; off <<= 1) v = fmaxf(v, __shfl_xor(v, off, 32));
    if (laneM == 0) out[(size_t)b * 1024 + mbase + r + hi * 8] = v;
  }
}

// ---------------------------------------------------------------------------
// plain 1x1 conv + bn + relu: (B,CIN,N) -> (B,128,N)
template <int CIN>
__global__ __launch_bounds__(256) void k_conv1x1(
    const float* __restrict__ xin, const float* __restrict__ W,
    const float* __restrict__ Wb, const float* __restrict__ Wg,
    const float* __restrict__ We, float* __restrict__ out) {
  constexpr int NK = CIN / 4;
  int b = blockIdx.y;
  int tid = threadIdx.x, wave = tid >> 5, lane = tid & 31, laneM = lane & 15, hi = lane >> 4;
  int row = wave * 16 + laneM;
  v2f a[NK];
#pragma unroll
  for (int s = 0; s < NK; ++s) {
    int ka = 4 * s + hi * 2;
    a[s].x = W[row * CIN + ka]; a[s].y = W[row * CIN + ka + 1];
  }
  float pb[8], pg[8], pe[8];
#pragma unroll
  for (int r = 0; r < 8; ++r) {
    int m = wave * 16 + r + hi * 8;
    pb[r] = Wb[m]; pg[r] = Wg[m]; pe[r] = We[m];
  }
  for (int t = 0; t < 8; ++t) {
    int n0 = blockIdx.x * 128 + t * 16;
    v8f acc = V8Z;
#pragma unroll
    for (int s = 0; s < NK; ++s) {
      int kb = 4 * s + hi * 2;
      v2f bv;
      bv.x = xin[(size_t)b * CIN * NPTS + kb * NPTS + n0 + laneM];
      bv.y = xin[(size_t)b * CIN * NPTS + (kb + 1) * NPTS + n0 + laneM];
      acc = wmma4(a[s], bv, acc);
    }
#pragma unroll
    for (int r = 0; r < 8; ++r) {
      int m = wave * 16 + r + hi * 8;
      out[(size_t)b * 128 * NPTS + m * NPTS + n0 + laneM] =
          fmaxf((acc[r] + pb[r]) * pg[r] + pe[r], 0.f);
    }
  }
}

// ---------------------------------------------------------------------------
// FC over batch 16 (single N-tile): out = [relu(bn(]x @ W^T + b[))]
template <bool BN>
__global__ __launch_bounds__(256) void k_fc(
    const float* __restrict__ xin, const float* __restrict__ W,
    const float* __restrict__ Wb, const float* __restrict__ Wg,
    const float* __restrict__ We, float* __restrict__ out, int K, int O) {
  int tid = threadIdx.x, wave = tid >> 5, lane = tid & 31, laneM = lane & 15, hi = lane >> 4;
  int row = blockIdx.x * 128 + wave * 16 + laneM;
  int NK = K / 4;
  v8f acc = V8Z;
  for (int s = 0; s < NK; ++s) {
    int ka = 4 * s + hi * 2;
    v2f av, bv;
    av.x = (row < O) ? W[row * K + ka] : 0.f;
    av.y = (row < O) ? W[row * K + ka + 1] : 0.f;
    bv.x = xin[laneM * K + ka];
    bv.y = xin[laneM * K + ka + 1];
    acc = wmma4(av, bv, acc);
  }
#pragma unroll
  for (int r = 0; r < 8; ++r) {
    int m = blockIdx.x * 128 + wave * 16 + r + hi * 8;
    if (m < O) {
      float v = acc[r] + Wb[m];
      if (BN) v = fmaxf(v * Wg[m] + We[m], 0.f);
      out[laneM * O + m] = v;
    }
  }
}

// ---------------------------------------------------------------------------
// apply 3x3 transform: xct[b,d,n] = sum_c xc[b,c,n] * T[b,c,d]
__global__ void k_transform(const float* __restrict__ xc, const float* __restrict__ t9,
                            float* __restrict__ xct) {
  int t = blockIdx.x * 256 + threadIdx.x;
  if (t >= B_SZ * NPTS) return;
  int b = t / NPTS, n = t % NPTS;
  float v0 = xc[b * 3072 + n], v1 = xc[b * 3072 + 1024 + n], v2 = xc[b * 3072 + 2048 + n];
  for (int d = 0; d < 3; ++d)
    xct[b * 3072 + d * 1024 + n] =
        v0 * t9[b * 9 + d] + v1 * t9[b * 9 + 3 + d] + v2 * t9[b * 9 + 6 + d];
}

// ---------------------------------------------------------------------------
// bilinear pool: p[c]=sqrt(sum_n a*b + 1e-5); l2norm over c; -> x5[b, q*128+c]
__global__ __launch_bounds__(128) void k_bilinear(const float* __restrict__ A,
                                                  const float* __restrict__ Bv,
                                                  float* __restrict__ x5, int q) {
  __shared__ float red[128];
  int b = blockIdx.x, c = threadIdx.x;
  const float* pa = A  + (size_t)b * 128 * NPTS + c * NPTS;
  const float* pq = Bv + (size_t)b * 128 * NPTS + c * NPTS;
  float s = 0.f;
  for (int n = 0; n < NPTS; ++n) s += pa[n] * pq[n];
  float p = sqrtf(s + 1e-5f);
  red[c] = p * p;
  __syncthreads();
  for (int st = 64; st > 0; st >>= 1) {
    if (c < st) red[c] += red[c + st];
    __syncthreads();
  }
  float nrm = fmaxf(sqrtf(red[0]), 1e-12f);
  x5[b * 384 + q * 128 + c] = p / nrm;
}

// ---------------------------------------------------------------------------
// margin head: logits + loss
__global__ __launch_bounds__(256) void k_head(const float* __restrict__ fea,
                                              const int* __restrict__ label,
                                              const float* __restrict__ cw,
                                              const float* __restrict__ mw,
                                              float* __restrict__ out) {
  __shared__ float sW[40 * 256];
  __shared__ float sF[16 * 256];
  __shared__ float sL[16 * 40];
  __shared__ float sLoss[16], sSmax[16];
  int tid = threadIdx.x;
  for (int i = tid; i < 40; i += 256) {
    float ss = 0.f;
    for (int c = 0; c < 256; ++c) { float v = mw[i * 256 + c]; ss += v * v; }
    float inv = 1.f / fmaxf(sqrtf(ss), 1e-12f);
    for (int c = 0; c < 256; ++c) sW[i * 256 + c] = mw[i * 256 + c] * inv;
  }
  for (int i = tid; i < 16; i += 256) {
    float ss = 0.f;
    for (int c = 0; c < 256; ++c) { float v = fea[i * 256 + c]; ss += v * v; }
    float inv = 1.f / fmaxf(sqrtf(ss), 1e-12f);
    for (int c = 0; c < 256; ++c) sF[i * 256 + c] = fea[i * 256 + c] * inv;
  }
  __syncthreads();
  for (int i = tid; i < 640; i += 256) {
    int b = i / 40, j = i % 40;
    float s = 0.f;
    for (int c = 0; c < 256; ++c) s += sF[b * 256 + c] * sW[j * 256 + c];
    sL[i] = s; out[i] = s;
  }
  __syncthreads();
  if (tid < 16) {
    int b = tid, lab = label[b];
    float sp = sL[b * 40 + lab];
    float zmax = -3.402823466e38f;
    for (int j = 0; j < 40; ++j) {
      if (j == lab) continue;
      float z = 9.6f * (sL[b * 40 + j] - sp + 0.83f * cw[j]);
      zmax = fmaxf(zmax, z);
    }
    float ssum = 0.f;
    for (int j = 0; j < 40; ++j) {
      if (j == lab) continue;
      float z = 9.6f * (sL[b * 40 + j] - sp + 0.83f * cw[j]);
      ssum += expf(z - zmax);
    }
    float lse = zmax + logf(ssum);
    sLoss[b] = fmaxf(lse, 0.f) + log1pf(expf(-fabsf(lse)));
    float mx = -3.402823466e38f;
    for (int j = 0; j < 40; ++j) {
      float v;
      if (j == lab) v = 0.f;
      else {
        float d = 0.f;
        for (int c = 0; c < 256; ++c) d += sW[lab * 256 + c] * sW[j * 256 + c];
        v = fminf(fmaxf(d, -1.f), 1.f);
      }
      mx = fmaxf(mx, v);
    }
    sSmax[b] = mx;
  }
  __syncthreads();
  if (tid == 0) {
    float ml = 0.f, ms = 0.f;
    for (int b = 0; b < 16; ++b) { ml += sLoss[b]; ms += sSmax[b]; }
    ml *= (1.f / 16.f); ms *= (1.f / 16.f);
    float xw = ms + 0.5f;
    float lw = fmaxf(xw, 0.f) + log1pf(expf(-fabsf(xw)));
    out[640] = ml + 10.f * lw;
  }
}

// ---------------------------------------------------------------------------
extern "C" void kernel_launch(void* const* d_in, const int* in_sizes, int n_in,
                              void* d_out, int out_size, void* d_ws, size_t ws_size,
                              hipStream_t stream) {
  (void)in_sizes; (void)n_in; (void)out_size; (void)ws_size;
  int p = 0;
  const float* X   = (const float*)d_in[p++];
  const int*   LAB = (const int*)d_in[p++];
  const float* CW  = (const float*)d_in[p++];
  const float* tn[5][4];                 // tnet c1,c2,c3,f1,f2  {W,b,g,be}
  for (int l = 0; l < 5; ++l) for (int q = 0; q < 4; ++q) tn[l][q] = (const float*)d_in[p++];
  const float* f3W = (const float*)d_in[p++];
  const float* f3b = (const float*)d_in[p++];
  const float* cv[10][4];                // convs c1,c2,c3,c4,c21,c22,c31,c32,c41,c42
  for (int l = 0; l < 10; ++l) for (int q = 0; q < 4; ++q) cv[l][q] = (const float*)d_in[p++];
  const float* ff[2][4];                 // fcs f1,f2
  for (int l = 0; l < 2; ++l) for (int q = 0; q < 4; ++q) ff[l][q] = (const float*)d_in[p++];
  const float* MW = (const float*)d_in[p++];

  float* ws = (float*)d_ws;
  size_t o = 0;
  float* xc   = ws + o; o += 3 * B_SZ * NPTS;
  float* xct  = ws + o; o += 3 * B_SZ * NPTS;
  int*   idxb = (int*)(ws + o); o += (size_t)B_SZ * NPTS * KNN;
  float* big0 = ws + o; o += (size_t)B_SZ * 128 * NPTS;
  float* big1 = ws + o; o += (size_t)B_SZ * 128 * NPTS;
  float* x1b  = ws + o; o += (size_t)B_SZ * 64 * NPTS;
  float* x2b  = ws + o; o += (size_t)B_SZ * 64 * NPTS;
  float* x3b  = ws + o; o += (size_t)B_SZ * 64 * NPTS;
  float* x4b  = ws + o; o += (size_t)B_SZ * 128 * NPTS;
  float* xxb  = ws + o; o += (size_t)B_SZ * NPTS;
  float* poolb= ws + o; o += (size_t)B_SZ * 1024;
  float* fcA  = ws + o; o += B_SZ * 512;
  float* fcB  = ws + o; o += B_SZ * 256;
  float* t9b  = ws + o; o += B_SZ * 9;
  float* x5b  = ws + o; o += B_SZ * 384;
  float* hb   = ws + o; o += B_SZ * 512;
  float* feab = ws + o; o += B_SZ * 256;

  dim3 gEC(NPTS / 4, B_SZ);
  dim3 gKN(NPTS / 128, B_SZ);

  k_transpose<<<(B_SZ * NPTS + 255) / 256, 256, 0, stream>>>(X, xc);
  // T-Net branch
  k_knn<3><<<gKN, 128, 0, stream>>>(xc, idxb);
  k_tnet12<<<gEC, 256, 0, stream>>>(xc, idxb,
      tn[0][0], tn[0][1], tn[0][2], tn[0][3],
      tn[1][0], tn[1][1], tn[1][2], tn[1][3], big0);
  k_gemm_maxN<<<dim3(8, B_SZ), 256, 0, stream>>>(big0,
      tn[2][0], tn[2][1], tn[2][2], tn[2][3], poolb);
  k_fc<true><<<4, 256, 0, stream>>>(poolb, tn[3][0], tn[3][1], tn[3][2], tn[3][3], fcA, 1024, 512);
  k_fc<true><<<2, 256, 0, stream>>>(fcA,   tn[4][0], tn[4][1], tn[4][2], tn[4][3], fcB, 512, 256);
  k_fc<false><<<1, 256, 0, stream>>>(fcB, f3W, f3b, f3b, f3b, t9b, 256, 9);
  k_transform<<<(B_SZ * NPTS + 255) / 256, 256, 0, stream>>>(xc, t9b, xct);
  // main edge-conv stack (64-dim kNNs on the matrix pipe)
  k_knn<3><<<gKN, 128, 0, stream>>>(xct, idxb);
  k_edgeconv<6, 64><<<gEC, 256, 0, stream>>>(xct, idxb,
      cv[0][0], cv[0][1], cv[0][2], cv[0][3], x1b);
  k_sqnorm64<<<(B_SZ * NPTS + 255) / 256, 256, 0, stream>>>(x1b, xxb);
  k_knn_wmma<<<gKN, 256, 0, stream>>>(x1b, xxb, idxb);
  k_edgeconv<128, 64><<<gEC, 256, 0, stream>>>(x1b, idxb,
      cv[1][0], cv[1][1], cv[1][2], cv[1][3], x2b);
  k_sqnorm64<<<(B_SZ * NPTS + 255) / 256, 256, 0, stream>>>(x2b, xxb);
  k_knn_wmma<<<gKN, 256, 0, stream>>>(x2b, xxb, idxb);
  k_edgeconv<128, 64><<<gEC, 256, 0, stream>>>(x2b, idxb,
      cv[2][0], cv[2][1], cv[2][2], cv[2][3], x3b);
  k_sqnorm64<<<(B_SZ * NPTS + 255) / 256, 256, 0, stream>>>(x3b, xxb);
  k_knn_wmma<<<gKN, 256, 0, stream>>>(x3b, xxb, idxb);
  k_edgeconv<128, 128><<<gEC, 256, 0, stream>>>(x3b, idxb,
      cv[3][0], cv[3][1], cv[3][2], cv[3][3], x4b);
  // bilinear pools
  dim3 gCV(8, B_SZ);
  k_conv1x1<64><<<gCV, 256, 0, stream>>>(x2b, cv[4][0], cv[4][1], cv[4][2], cv[4][3], big0);
  k_conv1x1<64><<<gCV, 256, 0, stream>>>(x3b, cv[5][0], cv[5][1], cv[5][2], cv[5][3], big1);
  k_bilinear<<<B_SZ, 128, 0, stream>>>(big0, big1, x5b, 0);
  k_conv1x1<64><<<gCV, 256, 0, stream>>>(x3b, cv[6][0], cv[6][1], cv[6][2], cv[6][3], big0);
  k_conv1x1<128><<<gCV, 256, 0, stream>>>(x4b, cv[7][0], cv[7][1], cv[7][2], cv[7][3], big1);
  k_bilinear<<<B_SZ, 128, 0, stream>>>(big0, big1, x5b, 1);
  k_conv1x1<128><<<gCV, 256, 0, stream>>>(x4b, cv[8][0], cv[8][1], cv[8][2], cv[8][3], big0);
  k_conv1x1<64><<<gCV, 256, 0, stream>>>(x1b, cv[9][0], cv[9][1], cv[9][2], cv[9][3], big1);
  k_bilinear<<<B_SZ, 128, 0, stream>>>(big0, big1, x5b, 2);
  // final FCs + head
  k_fc<true><<<4, 256, 0, stream>>>(x5b, ff[0][0], ff[0][1], ff[0][2], ff[0][3], hb, 384, 512);
  k_fc<true><<<2, 256, 0, stream>>>(hb,  ff[1][0], ff[1][1], ff[1][2], ff[1][3], feab, 512, 256);
  k_head<<<1, 256, 0, stream>>>(feab, LAB, CW, MW, (float*)d_out);
}